// BinaryHungarianMatcherV2_5952824672678
// MI455X (gfx1250) — compile-verified
//
#include <hip/hip_runtime.h>
#include <hip/hip_bf16.h>
#include <stdint.h>

// Problem constants (from reference setup_inputs)
#define B_TOTAL   32
#define Q_TOTAL   1800
#define T_TOTAL   500
#define QPB       20          // queries per block: 1800 / 20 = 90 blocks per batch, exact
#define NTHREADS  256         // 8 waves (wave32)

// TDM descriptor SGPR-group types
typedef unsigned int tdm_u32x4 __attribute__((ext_vector_type(4)));
typedef int          tdm_i32x8 __attribute__((ext_vector_type(8)));
typedef int          tdm_i32x4 __attribute__((ext_vector_type(4)));

// Discriminate toolchains: therock-10.0 headers (6-arg tensor builtin) ship the TDM header.
#if __has_include(<hip/amd_detail/amd_gfx1250_TDM.h>)
#define TDM_SIX_ARG 1
#else
#define TDM_SIX_ARG 0
#endif

__global__ __launch_bounds__(NTHREADS) void matcher_cost_kernel(
    const float* __restrict__ pred_logits,   // [B,Q,1]
    const float* __restrict__ pred_boxes,    // [B,Q,4] cxcywh
    const float* __restrict__ boxes_padded,  // [B,T,4] cxcywh
    const int*   __restrict__ num_boxes,     // [B]
    float*       __restrict__ out)           // [B,Q,T]
{
    // LDS staging: raw target boxes (TDM landing zone) + processed forms.
    __shared__ float4 rawT[T_TOTAL];   // cxcywh (8000 B, written by TDM)
    __shared__ float4 tXY [T_TOTAL];   // xyxy
    __shared__ float  tAr [T_TOTAL];   // area = w*h
    __shared__ float4 qRaw[QPB];       // cxcywh
    __shared__ float4 qXY [QPB];       // xyxy
    __shared__ float2 qMeta[QPB];      // (area, focal class cost)

    const int tid = threadIdx.x;
    const int b   = blockIdx.y;
    const int q0  = blockIdx.x * QPB;
    const int nb  = num_boxes[b];

    const float* tsrc = boxes_padded + (size_t)b * T_TOTAL * 4;

#if defined(__gfx1250__) && __has_builtin(__builtin_amdgcn_tensor_load_to_lds)
    // --- CDNA5 Tensor Data Mover: one-shot 8 KB global->LDS DMA, no VGPRs ---
    if (tid < 32) {  // one wave issues; EXEC is ignored by TENSOR_* ops
        const uint64_t gaddr = (uint64_t)(uintptr_t)tsrc;
        const uint32_t laddr = (uint32_t)(uintptr_t)(&rawT[0]); // flat LDS addr low 32 = LDS offset
        // D# group0: count=1, lds_addr, 57-bit global_addr, type=2 ("image")
        tdm_u32x4 g0;
        g0[0] = 1u;                                   // count=1, user descriptor
        g0[1] = laddr;
        g0[2] = (uint32_t)(gaddr & 0xFFFFFFFFu);
        g0[3] = ((uint32_t)(gaddr >> 32) & 0x01FFFFFFu) | (2u << 30);
        // D# group1: data_size=4B, 1-D tensor of 2000 elems, tile_dim0=2000
        tdm_i32x8 g1;
        g1[0] = (int)(2u << 16);                      // wg_mask=0 (not in cluster), data_size=2 (4B)
        g1[1] = (int)((2000u & 0xFFFFu) << 16);       // tensor_dim0[15:0] @ bits 63:48
        g1[2] = (int)(1u << 16);                      // tensor_dim0 hi=0; tensor_dim1=1 @ bits 95:80
        g1[3] = (int)((2000u & 0xFFFFu) << 16);       // tensor_dim1 hi=0; tile_dim0=2000 @ bits 127:112
        g1[4] = 0;                                    // tile_dim1=0 (unused), tile_dim2=0
        g1[5] = 2000;                                 // tensor_dim0_stride[31:0]
        g1[6] = (int)((2000u & 0xFFFFu) << 16);       // stride0 hi=0; tensor_dim1_stride[15:0]
        g1[7] = 0;
        tdm_i32x4 g2 = {0, 0, 0, 0};                  // dims 2+ unused
        tdm_i32x4 g3 = {0, 0, 0, 0};
#if TDM_SIX_ARG
        tdm_i32x8 g4 = {0, 0, 0, 0, 0, 0, 0, 0};      // extra group (clang-23 form), zero-filled
        __builtin_amdgcn_tensor_load_to_lds(g0, g1, g2, g3, g4, 0);
#else
        __builtin_amdgcn_tensor_load_to_lds(g0, g1, g2, g3, 0);
#endif
#if __has_builtin(__builtin_amdgcn_s_wait_tensorcnt)
        __builtin_amdgcn_s_wait_tensorcnt(0);
#endif
    }
#else
    // Fallback: cooperative vector copy (host pass / toolchains without the builtin)
    for (int t = tid; t < T_TOTAL; t += NTHREADS) {
        rawT[t] = *(const float4*)(tsrc + (size_t)t * 4);
    }
#endif
    __syncthreads();

    // ---- Preprocess targets: cxcywh -> xyxy + area (read one per lane later) ----
    for (int t = tid; t < T_TOTAL; t += NTHREADS) {
        const float4 r = rawT[t];
        const float hw = 0.5f * r.z, hh = 0.5f * r.w;
        tXY[t] = make_float4(r.x - hw, r.y - hh, r.x + hw, r.y + hh);
        tAr[t] = r.z * r.w;
    }
    // ---- Preprocess this block's queries: xyxy, area, focal class cost ----
    if (tid < QPB) {
        const int q = q0 + tid;
        const float4 pb = *(const float4*)(pred_boxes + ((size_t)b * Q_TOTAL + q) * 4);
        qRaw[tid] = pb;
        const float hw = 0.5f * pb.z, hh = 0.5f * pb.w;
        qXY[tid] = make_float4(pb.x - hw, pb.y - hh, pb.x + hw, pb.y + hh);
        const float s = pred_logits[(size_t)b * Q_TOTAL + q];
        // stable log_sigmoid(±s) = min(±s,0) - log1p(exp(-|s|))
        const float l1pe = __logf(1.0f + __expf(-fabsf(s)));
        const float ls_p = fminf(s, 0.0f) - l1pe;    // log sigmoid(s)
        const float ls_n = fminf(-s, 0.0f) - l1pe;   // log sigmoid(-s)
        const float p    = __builtin_amdgcn_rcpf(1.0f + __expf(-s));
        const float omp  = 1.0f - p;
        const float cc   = -0.25f * omp * omp * ls_p + 0.75f * p * p * ls_n;
        qMeta[tid] = make_float2(pb.z * pb.w, cc);
    }
    __syncthreads();

    // ---- Main compute: lanes over T (coalesced stores), loop over 20 queries ----
    float* const outBase = out + ((size_t)b * Q_TOTAL + q0) * T_TOTAL;

    #pragma unroll
    for (int tc = 0; tc < 2; ++tc) {
        const int  t      = tid + tc * NTHREADS;
        const bool tstore = (t < T_TOTAL);
        const int  tl     = tstore ? t : (T_TOTAL - 1);   // clamp for LDS reads
        const float4 tr = rawT[tl];
        const float4 tx = tXY [tl];
        const float  ta = tAr [tl];
        const bool   tvalid = (t < nb);

        #pragma unroll 4
        for (int i = 0; i < QPB; ++i) {
            const float4 qr = qRaw[i];
            const float4 qx = qXY [i];
            const float2 qm = qMeta[i];
            // L1 (cxcywh space); abs folds into VALU input modifiers
            const float l1 = fabsf(tr.x - qr.x) + fabsf(tr.y - qr.y)
                           + fabsf(tr.z - qr.z) + fabsf(tr.w - qr.w);
            // intersection
            const float ix0 = fmaxf(qx.x, tx.x);
            const float iy0 = fmaxf(qx.y, tx.y);
            const float ix1 = fminf(qx.z, tx.z);
            const float iy1 = fminf(qx.w, tx.w);
            const float iw  = fmaxf(ix1 - ix0, 0.0f);
            const float ih  = fmaxf(iy1 - iy0, 0.0f);
            const float inter = iw * ih;
            const float uni   = qm.x + ta - inter;
            // enclosing box
            const float ex0 = fminf(qx.x, tx.x);
            const float ey0 = fminf(qx.y, tx.y);
            const float ex1 = fmaxf(qx.z, tx.z);
            const float ey1 = fmaxf(qx.w, tx.w);
            const float ew  = fmaxf(ex1 - ex0, 0.0f);
            const float eh  = fmaxf(ey1 - ey0, 0.0f);
            const float ae  = ew * eh;
            // giou = iou - (ae - union)/ae  (two fast rcps)
            const float giou = inter * __builtin_amdgcn_rcpf(uni)
                             - (ae - uni) * __builtin_amdgcn_rcpf(ae);
            // C = 5*L1 + 2*cost_class - 2*giou; mask padded targets
            float c = fmaf(5.0f, l1, 2.0f * (qm.y - giou));
            c = tvalid ? c : 1.0e9f;
            if (tstore) {
                __builtin_nontemporal_store(c, outBase + (size_t)i * T_TOTAL + t);
            }
        }
    }
}

extern "C" void kernel_launch(void* const* d_in, const int* in_sizes, int n_in,
                              void* d_out, int out_size, void* d_ws, size_t ws_size,
                              hipStream_t stream) {
    const float* pred_logits  = (const float*)d_in[0];
    const float* pred_boxes   = (const float*)d_in[1];
    const float* boxes_padded = (const float*)d_in[2];
    const int*   num_boxes    = (const int*)d_in[3];
    float* out = (float*)d_out;

    const int B = in_sizes[3];                 // 32 batches (num_boxes is [B])
    dim3 grid(Q_TOTAL / QPB, B, 1);            // 90 x 32 = 2880 blocks
    hipLaunchKernelGGL(matcher_cost_kernel, grid, dim3(NTHREADS), 0, stream,
                       pred_logits, pred_boxes, boxes_padded, num_boxes, out);
}